// CLFQPSolver_72318659330100
// MI455X (gfx1250) — compile-verified
//
#include <hip/hip_runtime.h>
#include <stdint.h>

// CLF-QP batched solver for MI455X (gfx1250).
//   inputs : L_f_V [N] f32, L_g_V [N,32] f32, V [N] f32
//   outputs: u [N,32] f32 then r [N] f32 (concatenated in d_out)
//
// Memory-bound streaming kernel: 268 B/row * 1M rows = 268 MB @ 23.3 TB/s
// => ~11.5us floor. Compute reduced to ~200 VALU ops/row via the identity
//   a*clip(-0.5*nu*a, -5, 5) = max(-0.5*nu*a^2, -5*|a|)
// and the observation that for max|a_i| <= 2*UB/LAM the reference's
// bisection+Newton is exactly the closed form nu = 2b / sum(a^2).
// L_g_V is streamed through LDS with gfx1250 async global->LDS DMA
// (non-temporal: touched exactly once), double-buffered and ordered purely
// with s_wait_asynccnt (no barriers: each thread consumes only the LDS
// slice it DMA'd itself). All other traffic is non-temporal too - nothing
// in this kernel is ever re-read, so L2 retention is pure thrash.

#define A_DIM 32
#define BLOCK 256
#define ROW_DW 36                    // 32 data dwords + 4 pad dwords (bank-conflict relief)
#define LDS_FLOATS (2 * BLOCK * ROW_DW)

typedef float v4f __attribute__((ext_vector_type(4)));

__device__ __forceinline__ float clamp5(float x) {
    return fminf(fmaxf(x, -5.0f), 5.0f);
}

// Issue 8 async b128 loads: one 128-byte row of L_g_V -> this thread's LDS slice.
// GVS addressing: mem = SADDR(64-bit base) + VADDR(32-bit per-lane offset).
// Non-temporal: the row is consumed once from LDS and never revisited.
__device__ __forceinline__ void issue_async_row(uint32_t ldsByte, uint32_t gByteOff,
                                                uint64_t gBase) {
#pragma unroll
    for (int j = 0; j < 8; ++j) {
        asm volatile("global_load_async_to_lds_b128 %0, %1, %2 th:TH_LOAD_NT"
                     :: "v"(ldsByte + (uint32_t)(16 * j)),
                        "v"(gByteOff + (uint32_t)(16 * j)),
                        "s"(gBase)
                     : "memory");
    }
}

__global__ __launch_bounds__(BLOCK)
void clfqp_solver(const float* __restrict__ Lf,
                  const float* __restrict__ Lg,
                  const float* __restrict__ Vv,
                  float* __restrict__ u_out,
                  float* __restrict__ r_out,
                  int n, int nTiles) {
    __shared__ float lds[LDS_FLOATS];
    const int tid = (int)threadIdx.x;
    const uint64_t gBase = (uint64_t)(uintptr_t)Lg;

    float* mySlice0 = &lds[tid * ROW_DW];
    float* mySlice1 = &lds[(BLOCK + tid) * ROW_DW];
    // Low 32 bits of the generic (flat) pointer to LDS == LDS byte address.
    const uint32_t myLds0 = (uint32_t)(uintptr_t)mySlice0;
    const uint32_t myLds1 = (uint32_t)(uintptr_t)mySlice1;

    const int t0 = (int)blockIdx.x;
    const int stride = (int)gridDim.x;

    // Prologue: stage first tile into buffer 0 (clamp OOB rows to a valid row
    // so the per-wave ASYNCcnt bookkeeping stays uniform).
    {
        int row = t0 * BLOCK + tid;
        int rc = row < n ? row : (n - 1);
        issue_async_row(myLds0, (uint32_t)rc * (uint32_t)(A_DIM * 4), gBase);
    }

    int buf = 0;
    for (int t = t0; t < nTiles; t += stride, buf ^= 1) {
        const int tn = t + stride;
        if (tn < nTiles) {
            int nrow = tn * BLOCK + tid;
            int nrc = nrow < n ? nrow : (n - 1);
            issue_async_row(buf ? myLds0 : myLds1,
                            (uint32_t)nrc * (uint32_t)(A_DIM * 4), gBase);
            // 8 new loads outstanding; <=8 means the previous tile's 8 are done.
            asm volatile("s_wait_asynccnt 0x8" ::: "memory");
        } else {
            asm volatile("s_wait_asynccnt 0x0" ::: "memory");
        }

        const int row = t * BLOCK + tid;
        if (row < n) {
            const float* ap = buf ? mySlice1 : mySlice0;
            float a[A_DIM];
#pragma unroll
            for (int j = 0; j < 8; ++j) {
                v4f v = *(const v4f*)(ap + 4 * j);        // ds_load_b128
                a[4 * j + 0] = v.x; a[4 * j + 1] = v.y;
                a[4 * j + 2] = v.z; a[4 * j + 3] = v.w;
            }

            const float b = __builtin_nontemporal_load(Lf + row) +
                            __builtin_nontemporal_load(Vv + row);   // C = 1
            float S = 0.0f, amax = 0.0f, gl = 0.0f;
#pragma unroll
            for (int i = 0; i < A_DIM; ++i) {
                float ai = a[i];
                float s  = ai * ai;
                float ab = fabsf(ai);
                S    += s;
                amax  = fmaxf(amax, ab);
                gl   += fmaxf(-0.5f * s, -5.0f * ab);   // a*clip(-0.5*LAM*a) term
            }
            const float g_lam = b + gl;

            float nu, r;
            if (b <= 0.0f) {
                nu = 0.0f; r = 0.0f;
            } else if (g_lam > 0.0f) {
                nu = 1.0f; r = g_lam;                    // LAM = 1
            } else {
                r = 0.0f;
                if (0.5f * amax <= 5.0f) {
                    // No element can saturate for nu in [0,1]: g is linear and
                    // bisection + Newton reduce exactly to the closed-form root.
                    nu = (2.0f * b) / S;
                } else {
                    // Cold path: faithful 60-step bisection + Newton.
                    float lo = 0.0f, hi = 1.0f;
#pragma unroll 1
                    for (int it = 0; it < 60; ++it) {
                        float mid = 0.5f * (lo + hi);
                        float g = b;
#pragma unroll
                        for (int i = 0; i < A_DIM; ++i) {
                            float ai = a[i];
                            g += fmaxf(-0.5f * mid * ai * ai, -5.0f * fabsf(ai));
                        }
                        if (g > 0.0f) lo = mid; else hi = mid;
                    }
                    float nu0 = 0.5f * (lo + hi);
                    float g0 = b, sdg = 0.0f;
#pragma unroll
                    for (int i = 0; i < A_DIM; ++i) {
                        float ai = a[i];
                        float ab = fabsf(ai);
                        g0 += fmaxf(-0.5f * nu0 * ai * ai, -5.0f * ab);
                        if (0.5f * nu0 * ab < 5.0f)      // u0 strictly interior
                            sdg += ai * ai;
                    }
                    float dg = -0.5f * sdg;
                    if (!(dg < 0.0f)) dg = -1.0f;        // degenerate guard
                    nu = nu0 - g0 / dg;
                }
            }

            v4f* uo = (v4f*)(u_out + (size_t)row * A_DIM);
#pragma unroll
            for (int j = 0; j < 8; ++j) {
                v4f o;
                o.x = clamp5(-0.5f * nu * a[4 * j + 0]);
                o.y = clamp5(-0.5f * nu * a[4 * j + 1]);
                o.z = clamp5(-0.5f * nu * a[4 * j + 2]);
                o.w = clamp5(-0.5f * nu * a[4 * j + 3]);
                __builtin_nontemporal_store(o, uo + j);   // global_store_b128 NT
            }
            __builtin_nontemporal_store(r, r_out + row);
        }
    }
}

extern "C" void kernel_launch(void* const* d_in, const int* in_sizes, int n_in,
                              void* d_out, int out_size, void* d_ws, size_t ws_size,
                              hipStream_t stream) {
    (void)n_in; (void)out_size; (void)d_ws; (void)ws_size;
    const float* Lf = (const float*)d_in[0];   // L_f_V [N]
    const float* Lg = (const float*)d_in[1];   // L_g_V [N,32]
    const float* Vv = (const float*)d_in[2];   // V     [N]
    const int n = in_sizes[0];
    if (n <= 0) return;

    float* u = (float*)d_out;
    float* r = u + (size_t)n * A_DIM;

    const int nTiles = (n + BLOCK - 1) / BLOCK;
    int grid = nTiles < 1024 ? nTiles : 1024;  // persistent blocks => per-block
                                               // async pipeline depth nTiles/grid
    clfqp_solver<<<grid, BLOCK, 0, stream>>>(Lf, Lg, Vv, u, r, n, nTiles);
}